// GCN_94489280548
// MI455X (gfx1250) — compile-verified
//
#include <hip/hip_runtime.h>
#include <cstddef>

typedef __attribute__((ext_vector_type(2))) float v2f;
typedef __attribute__((ext_vector_type(8))) float v8f;

// ---------------------------------------------------------------------------
// GEMM: H[N x NCOLS] = act(X[N x 128]) @ W[128 x NCOLS]
// W staged in LDS in k-pair-interleaved layout: element (k, col) lives at
// lds[(k>>1)*S + col*2 + (k&1)], so each WMMA B-fragment {W[k][c], W[k+1][c]}
// is one aligned ds_load_b64. S chosen so the two half-waves hit disjoint
// bank sets. One wave computes a 16-row x NCOLS slab via f32 WMMA 16x16x4.
// ---------------------------------------------------------------------------
template <int NCOLS, bool RELU_IN>
__global__ __launch_bounds__(256) void gemm_wmma_k(const float* __restrict__ X,
                                                   const float* __restrict__ W,
                                                   float* __restrict__ H,
                                                   int ntiles) {
    constexpr int S = (NCOLS == 128) ? 288 : NCOLS * 2;  // dword stride per k-pair
    __shared__ float ldsW[64 * S];

    // cooperative stage of W[128 x NCOLS] into pair-interleaved LDS
    for (int idx = threadIdx.x; idx < 128 * NCOLS; idx += 256) {
        const int r = idx / NCOLS;       // k
        const int c = idx % NCOLS;       // col
        ldsW[(r >> 1) * S + c * 2 + (r & 1)] = W[(size_t)r * NCOLS + c];
    }
    __syncthreads();

    const int lane = threadIdx.x & 31;
    const int wave = threadIdx.x >> 5;
    const int tile = blockIdx.x * 8 + wave;
    if (tile >= ntiles) return;               // wave-uniform: EXEC stays all-1s
    const int row0 = tile * 16;
    const int m    = lane & 15;
    const int hi   = lane >> 4;               // 0 or 1 (half-wave)
    constexpr int NT = NCOLS / 16;

    v8f acc[NT] = {};
    const float* xrow = X + (size_t)(row0 + m) * 128;

    for (int k = 0; k < 128; k += 4) {
        // A fragment: lane holds A[m][k + 2*hi + {0,1}]
        float a0 = xrow[k + 2 * hi];
        float a1 = xrow[k + 2 * hi + 1];
        if (RELU_IN) { a0 = fmaxf(a0, 0.f); a1 = fmaxf(a1, 0.f); }
        v2f a; a[0] = a0; a[1] = a1;
        const float* wrow = ldsW + ((k >> 1) + hi) * S;
#pragma unroll
        for (int t = 0; t < NT; ++t) {
            // B fragment: contiguous pair {W[k+2hi][t*16+m], W[k+2hi+1][t*16+m]}
            const v2f b = *(const v2f*)(wrow + (t * 16 + m) * 2);
            acc[t] = __builtin_amdgcn_wmma_f32_16x16x4_f32(
                false, a, false, b, (short)0, acc[t], false, false);
        }
    }

#pragma unroll
    for (int t = 0; t < NT; ++t) {
        float* o = H + (size_t)row0 * NCOLS + t * 16 + m;
#pragma unroll
        for (int v = 0; v < 8; ++v)
            o[(size_t)(v + 8 * hi) * NCOLS] = acc[t][v];
    }
}

// ---------------------------------------------------------------------------
// Degree / normalization / CSR build
// ---------------------------------------------------------------------------
__global__ __launch_bounds__(256) void k_zero_i(int* p, int n) {
    int i = blockIdx.x * blockDim.x + threadIdx.x;
    if (i < n) p[i] = 0;
}

__global__ __launch_bounds__(256) void k_count_i(const int* __restrict__ dst,
                                                 int* __restrict__ deg, int e) {
    int i = blockIdx.x * blockDim.x + threadIdx.x;
    if (i < e) atomicAdd(&deg[dst[i]], 1);
}

__global__ __launch_bounds__(256) void k_dis(const int* __restrict__ deg,
                                             float* __restrict__ dis, int n) {
    int i = blockIdx.x * blockDim.x + threadIdx.x;
    if (i < n) dis[i] = rsqrtf((float)deg[i] + 1.f);  // +1 = self-loop
}

// two-level exclusive scan over deg -> rowptr (chunk = 256 threads x 8)
__global__ __launch_bounds__(256) void k_scan1(const int* __restrict__ d,
                                               int* __restrict__ exc,
                                               int* __restrict__ bsum, int n) {
    __shared__ int sh[256];
    const int tid  = threadIdx.x;
    const int base = blockIdx.x * 2048 + tid * 8;
    int v[8]; int s = 0;
#pragma unroll
    for (int j = 0; j < 8; ++j) { v[j] = (base + j < n) ? d[base + j] : 0; s += v[j]; }
    sh[tid] = s;
    __syncthreads();
    for (int off = 1; off < 256; off <<= 1) {
        int t = (tid >= off) ? sh[tid - off] : 0;
        __syncthreads();
        if (tid >= off) sh[tid] += t;
        __syncthreads();
    }
    const int excl = sh[tid] - s;
    if (tid == 255) bsum[blockIdx.x] = sh[255];
    int run = excl;
#pragma unroll
    for (int j = 0; j < 8; ++j) {
        if (base + j < n) exc[base + j] = run;
        run += v[j];
    }
}

__global__ void k_scan2(int* bsum, int nb) {
    if (threadIdx.x == 0 && blockIdx.x == 0) {
        int run = 0;
        for (int i = 0; i < nb; ++i) { int t = bsum[i]; bsum[i] = run; run += t; }
    }
}

__global__ __launch_bounds__(256) void k_scan3(int* __restrict__ exc,
                                               const int* __restrict__ bsum,
                                               int* __restrict__ cursor, int n) {
    int i = blockIdx.x * blockDim.x + threadIdx.x;
    if (i < n) {
        int r = exc[i] + bsum[i >> 11];
        exc[i] = r;
        cursor[i] = r;
    }
}

// bucket edges by dst; meta = {src, bits(dis[src])}
__global__ __launch_bounds__(256) void k_bucket(const int* __restrict__ src,
                                                const int* __restrict__ dst,
                                                const float* __restrict__ dis,
                                                int* __restrict__ cursor,
                                                int2* __restrict__ meta, int e) {
    int i = blockIdx.x * blockDim.x + threadIdx.x;
    if (i >= e) return;
    int s = src[i];
    int pos = atomicAdd(&cursor[dst[i]], 1);
    meta[pos] = make_int2(s, __float_as_int(dis[s]));
}

// ---------------------------------------------------------------------------
// Gather aggregation (no float atomics): one wave per dst node.
// B[i][:] = bias + dis[i] * ( H[i][:]*dis[i] + sum_j H[src_j][:]*dis[src_j] )
// ---------------------------------------------------------------------------
__global__ __launch_bounds__(256) void k_gather128(const float* __restrict__ H,
                                                   const int* __restrict__ rowptr,
                                                   const int2* __restrict__ meta,
                                                   const float* __restrict__ dis,
                                                   const float* __restrict__ bias,
                                                   float* __restrict__ B,
                                                   int n, int e) {
    const int node = (blockIdx.x * blockDim.x + threadIdx.x) >> 5;
    const int lane = threadIdx.x & 31;
    if (node >= n) return;
    const int beg = rowptr[node];
    const int end = (node + 1 < n) ? rowptr[node + 1] : e;
    const float di = dis[node];
    const int c = lane * 4;

    const float4 hs = *(const float4*)(H + (size_t)node * 128 + c);
    float4 acc;  // self-loop term
    acc.x = hs.x * di; acc.y = hs.y * di; acc.z = hs.z * di; acc.w = hs.w * di;

    for (int p = beg; p < end; ++p) {
        const int2 m = meta[p];
        const float w = __int_as_float(m.y);
        const float4 v = *(const float4*)(H + (size_t)m.x * 128 + c);
        acc.x = fmaf(v.x, w, acc.x); acc.y = fmaf(v.y, w, acc.y);
        acc.z = fmaf(v.z, w, acc.z); acc.w = fmaf(v.w, w, acc.w);
    }
    const float4 bb = *(const float4*)(bias + c);
    float4 r;
    r.x = fmaf(di, acc.x, bb.x); r.y = fmaf(di, acc.y, bb.y);
    r.z = fmaf(di, acc.z, bb.z); r.w = fmaf(di, acc.w, bb.w);
    *(float4*)(B + (size_t)node * 128 + c) = r;
}

// 16-col gather: 4 lanes x float4 per edge, 8 edges in flight per wave,
// cross-lane reduction via shfl_xor, lanes 0-3 write.
__global__ __launch_bounds__(256) void k_gather16(const float* __restrict__ H,
                                                  const int* __restrict__ rowptr,
                                                  const int2* __restrict__ meta,
                                                  const float* __restrict__ dis,
                                                  const float* __restrict__ bias,
                                                  float* __restrict__ out,
                                                  int n, int e) {
    const int node = (blockIdx.x * blockDim.x + threadIdx.x) >> 5;
    const int lane = threadIdx.x & 31;
    if (node >= n) return;
    const int beg = rowptr[node];
    const int end = (node + 1 < n) ? rowptr[node + 1] : e;
    const float di = dis[node];
    const int g  = lane >> 2;        // edge sub-slot 0..7
    const int q  = (lane & 3) * 4;   // float4 column chunk

    float4 acc = make_float4(0.f, 0.f, 0.f, 0.f);
    if (g == 0) {                    // self-loop term once
        const float4 hs = *(const float4*)(H + (size_t)node * 16 + q);
        acc.x = hs.x * di; acc.y = hs.y * di; acc.z = hs.z * di; acc.w = hs.w * di;
    }
    for (int p = beg + g; p < end; p += 8) {
        const int2 m = meta[p];
        const float w = __int_as_float(m.y);
        const float4 v = *(const float4*)(H + (size_t)m.x * 16 + q);
        acc.x = fmaf(v.x, w, acc.x); acc.y = fmaf(v.y, w, acc.y);
        acc.z = fmaf(v.z, w, acc.z); acc.w = fmaf(v.w, w, acc.w);
    }
#pragma unroll
    for (int mask = 4; mask <= 16; mask <<= 1) {
        acc.x += __shfl_xor(acc.x, mask, 32);
        acc.y += __shfl_xor(acc.y, mask, 32);
        acc.z += __shfl_xor(acc.z, mask, 32);
        acc.w += __shfl_xor(acc.w, mask, 32);
    }
    if (lane < 4) {
        const float4 bb = *(const float4*)(bias + q);
        float4 r;
        r.x = fmaf(di, acc.x, bb.x); r.y = fmaf(di, acc.y, bb.y);
        r.z = fmaf(di, acc.z, bb.z); r.w = fmaf(di, acc.w, bb.w);
        *(float4*)(out + (size_t)node * 16 + q) = r;
    }
}

// ---------------------------------------------------------------------------
extern "C" void kernel_launch(void* const* d_in, const int* in_sizes, int n_in,
                              void* d_out, int out_size, void* d_ws, size_t ws_size,
                              hipStream_t stream) {
    const float* x   = (const float*)d_in[0];
    const int*   ei  = (const int*)d_in[1];
    const float* W1  = (const float*)d_in[2];
    const float* b1  = (const float*)d_in[3];
    const float* W2  = (const float*)d_in[4];
    const float* b2  = (const float*)d_in[5];
    const float* W3  = (const float*)d_in[6];
    const float* b3  = (const float*)d_in[7];
    float*       out = (float*)d_out;

    const int N = in_sizes[0] / 128;
    const int E = in_sizes[1] / 2;
    const int* src = ei;        // edge_index[0]
    const int* dst = ei + E;    // edge_index[1]

    // workspace layout
    float* A      = (float*)d_ws;               // N x 128
    float* B      = A + (size_t)N * 128;        // N x 128
    int2*  meta   = (int2*)(B + (size_t)N * 128);   // E  (8B aligned)
    float* dis    = (float*)(meta + E);         // N
    int*   degi   = (int*)(dis + N);            // N
    int*   rowptr = degi + N;                   // N
    int*   cursor = rowptr + N;                 // N
    int*   bsum   = cursor + N;                 // nb (<=64)

    const int TPB = 256;
    auto cdiv = [](long long a, long long b) { return (int)((a + b - 1) / b); };
    const int ntiles = (N + 15) / 16;           // 6250 for N=100000
    const int nb     = cdiv(N, 2048);

    // degrees -> dis, CSR build
    k_zero_i <<<cdiv(N, TPB), TPB, 0, stream>>>(degi, N);
    k_count_i<<<cdiv(E, TPB), TPB, 0, stream>>>(dst, degi, E);
    k_dis    <<<cdiv(N, TPB), TPB, 0, stream>>>(degi, dis, N);
    k_scan1  <<<nb, TPB, 0, stream>>>(degi, rowptr, bsum, N);
    k_scan2  <<<1, 32, 0, stream>>>(bsum, nb);
    k_scan3  <<<cdiv(N, TPB), TPB, 0, stream>>>(rowptr, bsum, cursor, N);
    k_bucket <<<cdiv(E, TPB), TPB, 0, stream>>>(src, dst, dis, cursor, meta, E);

    // Layer 1
    gemm_wmma_k<128, false><<<cdiv(ntiles, 8), TPB, 0, stream>>>(x, W1, A, ntiles);
    k_gather128<<<cdiv((long long)N * 32, TPB), TPB, 0, stream>>>(A, rowptr, meta, dis, b1, B, N, E);
    // Layer 2
    gemm_wmma_k<128, true><<<cdiv(ntiles, 8), TPB, 0, stream>>>(B, W2, A, ntiles);
    k_gather128<<<cdiv((long long)N * 32, TPB), TPB, 0, stream>>>(A, rowptr, meta, dis, b2, B, N, E);
    // Layer 3
    gemm_wmma_k<16, true><<<cdiv(ntiles, 8), TPB, 0, stream>>>(B, W3, A, ntiles);
    k_gather16<<<cdiv((long long)N * 32, TPB), TPB, 0, stream>>>(A, rowptr, meta, dis, b3, out, N, E);
}